// LogSigRNN_41515153883174
// MI455X (gfx1250) — compile-verified
//
#include <hip/hip_runtime.h>
#include <math.h>

#define IN_CH   64
#define SEGS    50
#define LSEG    4
#define TLEN    200
#define VDIM    25
#define NPAIR   2016            // 64*63/2
#define FDIM    2144            // 64 + 2016 + 64
#define HID     128
#define GDIM    512             // 4*HID
#define NSEQ    400             // 16*25
#define MROWS   (NSEQ*SEGS)     // 20000

typedef __attribute__((ext_vector_type(2))) float v2f;
typedef __attribute__((ext_vector_type(4))) float v4f;
typedef __attribute__((ext_vector_type(8))) float v8f;

// ---------------------------------------------------------------------------
// Kernel 1: segment log-signature features.  One block per (seq, segment).
// Emits seq[m][0:64]=lvl1, [64:2080]=levy area (triu), [2080:2144]=start.
// ---------------------------------------------------------------------------
__global__ void logsig_kernel(const float* __restrict__ x,   // (16,64,200,25)
                              float* __restrict__ seq)       // (20000, 2144)
{
    const int m = blockIdx.x;            // m = n*50 + s
    const int n = m / SEGS;
    const int s = m % SEGS;
    const int b = n / VDIM;
    const int v = n % VDIM;
    const int tid = threadIdx.x;

    __shared__ float p[LSEG][IN_CH];
    __shared__ float dv[LSEG - 1][IN_CH];
    __shared__ float ic[LSEG - 1][IN_CH];

    {   // 4*64 = 256 loads, one per thread.  x[b][c][s*4+t][v]
        const int t = tid / IN_CH, c = tid % IN_CH;
        p[t][c] = x[(((size_t)b * IN_CH + c) * TLEN + (s * LSEG + t)) * VDIM + v];
    }
    __syncthreads();
    if (tid < 3 * IN_CH) {
        const int t = tid / IN_CH, c = tid % IN_CH;
        ic[t][c] = p[t + 1][c] - p[t][c];
        dv[t][c] = p[t][c] - p[0][c];
    }
    __syncthreads();

    float* row = seq + (size_t)m * FDIM;
    if (tid < IN_CH) {
        row[tid] = p[LSEG - 1][tid] - p[0][tid];   // level-1
        row[IN_CH + NPAIR + tid] = p[0][tid];      // start
    }
    for (int k = tid; k < NPAIR; k += blockDim.x) {
        // invert row-major triu(k=1) index -> (i, j)
        int i = 0, off = 0;
        while (k >= off + (IN_CH - 1 - i)) { off += IN_CH - 1 - i; ++i; }
        const int j = i + 1 + (k - off);
        float a = 0.f;
        #pragma unroll
        for (int t = 0; t < 3; ++t)
            a += dv[t][i] * ic[t][j] - dv[t][j] * ic[t][i];
        row[IN_CH + k] = 0.5f * a;
    }
}

// ---------------------------------------------------------------------------
// Kernel 2: x_proj = seq @ W_ih^T + b_ih + b_hh   via V_WMMA_F32_16X16X4_F32.
// Block: 256 threads (8 waves), 32x128 output tile, K staged in 32-chunks.
//   A tile : [row][k], stride 36 (16B aligned rows, bank-clean fragments)
//            staged as ONE b128 global load + ONE b128 LDS store per thread.
//   B tile : quad-interleaved [k4][n]{k..k+3}; staged as b128/b128; WMMA
//            fragment (ka, ka+1) is one aligned ds_load_b64.
// ---------------------------------------------------------------------------
#define BK 32
#define ASTR 36
__global__ void xproj_gemm(const float* __restrict__ A,    // (20000, 2144)
                           const float* __restrict__ W,    // (512, 2144)
                           const float* __restrict__ bih,  // (512)
                           const float* __restrict__ bhh,  // (512)
                           float* __restrict__ Cout)       // (20000, 512)
{
    __shared__ float As[32][ASTR];
    __shared__ float Bs[(BK / 4) * 128 * 4];    // [k4][n][4]

    const int m0 = blockIdx.x * 32;
    const int n0 = blockIdx.y * 128;
    const int tid  = threadIdx.x;
    const int wave = tid >> 5;
    const int lane = tid & 31;
    const int wm = (wave >> 2) * 16;     // wave row offset within block tile
    const int wn = (wave & 3) * 32;      // wave col offset within block tile
    const int lm = lane & 15;
    const int hi = lane >> 4;

    // staging coordinates (loop-invariant)
    const int ar = tid >> 3;             // A row 0..31
    const int ak = (tid & 7) << 2;       // A k   0,4,..,28

    v8f acc0 = {}, acc1 = {};

    for (int k0 = 0; k0 < FDIM; k0 += BK) {
        // stage A tile (32x32): one float4 per thread
        *(v4f*)&As[ar][ak] = *(const v4f*)&A[(size_t)(m0 + ar) * FDIM + k0 + ak];
        // stage B tile (quad-interleaved): 4 float4 per thread
        #pragma unroll
        for (int q = 0; q < 4; ++q) {
            const int idx = tid + 256 * q;     // 0..1023 float4s
            const int nn = idx >> 3, k4 = idx & 7;
            *(v4f*)&Bs[(k4 * 128 + nn) * 4] =
                *(const v4f*)&W[(size_t)(n0 + nn) * FDIM + k0 + 4 * k4];
        }
        if (k0 + BK < FDIM)  // hint next A chunk into cache (global_prefetch_b8)
            __builtin_prefetch(&A[(size_t)(m0 + ar) * FDIM + k0 + BK + ak], 0, 1);
        __syncthreads();

        #pragma unroll
        for (int kk = 0; kk < BK; kk += 4) {
            const int ka = kk + 2 * hi;   // lanes 0-15: K=kk,kk+1; lanes 16-31: K=kk+2,kk+3
            const v2f a  = *(const v2f*)&As[wm + lm][ka];
            const v2f b0 = *(const v2f*)&Bs[((ka >> 2) * 128 + wn + lm)      * 4 + (ka & 3)];
            const v2f b1 = *(const v2f*)&Bs[((ka >> 2) * 128 + wn + 16 + lm) * 4 + (ka & 3)];
            acc0 = __builtin_amdgcn_wmma_f32_16x16x4_f32(false, a, false, b0,
                                                         (short)0, acc0, false, false);
            acc1 = __builtin_amdgcn_wmma_f32_16x16x4_f32(false, a, false, b1,
                                                         (short)0, acc1, false, false);
        }
        __syncthreads();
    }

    // epilogue: D layout  VGPR r, lane l -> M = r + 8*(l>>4), N = l&15
    #pragma unroll
    for (int r = 0; r < 8; ++r) {
        const size_t mrow = (size_t)(m0 + wm + r + 8 * hi) * GDIM;
        const int n  = n0 + wn + lm;
        const int n2 = n + 16;
        Cout[mrow + n]  = acc0[r] + bih[n]  + bhh[n];
        Cout[mrow + n2] = acc1[r] + bih[n2] + bhh[n2];
    }
}

// ---------------------------------------------------------------------------
// Kernel 3: LSTM recurrence.  25 blocks x 16 sequences.  W_hh^T (256 KB)
// resident in CDNA5's 320KB LDS for all 50 steps; gates via WMMA f32.
// W_hh stored quad-interleaved: Wl[k4*512 + n][0..3] = Whh[n][4k4 .. 4k4+3];
// one-time fill is b128 global loads -> b128 LDS stores; each WMMA B
// fragment is one aligned ds_load_b64.
// ---------------------------------------------------------------------------
#define HSTR 130   // padded h stride (even -> aligned v2f; odd bank group)

__global__ void lstm_kernel(const float* __restrict__ xproj,  // (20000, 512)
                            const float* __restrict__ Whh,    // (512, 128)
                            float* __restrict__ out)          // (16,128,50,25)
{
    extern __shared__ float sm[];
    float* Wl = sm;                        // [32][512][4] quads = 256 KB
    float* gl = Wl + HID * GDIM;           // [16][512]   gate tiles
    float* hl = gl + 16 * GDIM;            // [16][HSTR]  hidden state
    float* cl = hl + 16 * HSTR;            // [16][128]   cell state

    const int tid  = threadIdx.x;
    const int wave = tid >> 5;
    const int lane = tid & 31;
    const int lm = lane & 15;
    const int hi = lane >> 4;
    const int n0 = blockIdx.x * 16;        // first sequence of this block

    // one-time: W_hh -> LDS as k-quads (coalesced b128 global reads)
    for (int i = tid; i < (HID / 4) * GDIM; i += 256) {
        const int k4 = i & 31, n = i >> 5;      // consecutive threads -> consecutive k4
        *(v4f*)&Wl[(k4 * GDIM + n) * 4] = *(const v4f*)&Whh[(size_t)n * HID + 4 * k4];
    }
    for (int i = tid; i < 16 * HSTR; i += 256) hl[i] = 0.f;
    for (int i = tid; i < 16 * HID;  i += 256) cl[i] = 0.f;
    __syncthreads();

    for (int t = 0; t < SEGS; ++t) {
        // ---- gates = xproj[:, t, :] + h @ W_hh^T  (per wave: 4 col tiles) ----
        v8f acc[4];
        #pragma unroll
        for (int j = 0; j < 4; ++j) {
            const int nn = wave * 64 + j * 16 + lm;
            #pragma unroll
            for (int r = 0; r < 8; ++r) {
                const int mrow = r + 8 * hi;
                acc[j][r] = xproj[((size_t)(n0 + mrow) * SEGS + t) * GDIM + nn];
            }
        }
        #pragma unroll 4
        for (int k = 0; k < HID; k += 4) {
            const int ka = k + 2 * hi;
            const v2f a = *(const v2f*)&hl[lm * HSTR + ka];
            #pragma unroll
            for (int j = 0; j < 4; ++j) {
                const int nn = wave * 64 + j * 16 + lm;
                const v2f b = *(const v2f*)&Wl[((ka >> 2) * GDIM + nn) * 4 + (ka & 3)];
                acc[j] = __builtin_amdgcn_wmma_f32_16x16x4_f32(false, a, false, b,
                                                               (short)0, acc[j], false, false);
            }
        }
        // spill gate tiles to LDS
        #pragma unroll
        for (int j = 0; j < 4; ++j) {
            const int nn = wave * 64 + j * 16 + lm;
            #pragma unroll
            for (int r = 0; r < 8; ++r)
                gl[(r + 8 * hi) * GDIM + nn] = acc[j][r];
        }
        __syncthreads();

        // ---- elementwise LSTM cell update: 16*128 cells, 8 per thread ----
        for (int i = tid; i < 16 * HID; i += 256) {
            const int mrow = i >> 7, hc = i & 127;
            const float* g = gl + mrow * GDIM;
            const float ig = g[hc], fg = g[HID + hc], gg = g[2 * HID + hc], og = g[3 * HID + hc];
            const float si = 1.f / (1.f + expf(-ig));
            const float sf = 1.f / (1.f + expf(-fg));
            const float so = 1.f / (1.f + expf(-og));
            const float cn = sf * cl[i] + si * tanhf(gg);
            const float hn = so * tanhf(cn);
            cl[i] = cn;
            hl[mrow * HSTR + hc] = hn;
            const int n = n0 + mrow;
            const int b = n / VDIM, v = n % VDIM;
            out[((size_t)(b * HID + hc) * SEGS + t) * VDIM + v] = hn;
        }
        __syncthreads();   // h/c ready before next step's WMMA reads
    }
}

// ---------------------------------------------------------------------------
extern "C" void kernel_launch(void* const* d_in, const int* in_sizes, int n_in,
                              void* d_out, int out_size, void* d_ws, size_t ws_size,
                              hipStream_t stream) {
    const float* x    = (const float*)d_in[0];   // (16,64,200,25)
    const float* Wih  = (const float*)d_in[1];   // (512,2144)
    const float* Whh  = (const float*)d_in[2];   // (512,128)
    const float* bih  = (const float*)d_in[3];   // (512)
    const float* bhh  = (const float*)d_in[4];   // (512)
    float* out = (float*)d_out;                  // (16,128,50,25)

    float* seq   = (float*)d_ws;                           // 20000*2144 floats
    float* xproj = seq + (size_t)MROWS * FDIM;             // 20000*512 floats

    logsig_kernel<<<MROWS, 256, 0, stream>>>(x, seq);

    dim3 g2(MROWS / 32, GDIM / 128);
    xproj_gemm<<<g2, 256, 0, stream>>>(seq, Wih, bih, bhh, xproj);

    const size_t lds = (size_t)(HID * GDIM + 16 * GDIM + 16 * HSTR + 16 * HID) * sizeof(float);
    lstm_kernel<<<NSEQ / 16, 256, lds, stream>>>(xproj, Whh, out);
}